// SimpleMambaClassifierLuenberger_14353780703723
// MI455X (gfx1250) — compile-verified
//
#include <hip/hip_runtime.h>
#include <hip/hip_bf16.h>
#include <math.h>

// ---- model dims ----
#define BB   8
#define LL   1024          // 32*32 tokens
#define DM   512           // d_model
#define DI   1024          // d_inner
#define DS   16            // d_state
#define DTR  32            // dt_rank
#define NL   4
#define NC   10
#define ALPHA_C 0.2f
#define LNEPS   1e-5f

typedef float v2f __attribute__((ext_vector_type(2)));
typedef float v8f __attribute__((ext_vector_type(8)));
typedef unsigned int u32x4 __attribute__((ext_vector_type(4)));
typedef int i32x4 __attribute__((ext_vector_type(4)));
typedef int i32x8 __attribute__((ext_vector_type(8)));

__device__ __forceinline__ float sigmoidf_(float x){ return 1.0f/(1.0f+__expf(-x)); }
__device__ __forceinline__ float siluf_(float x){ return x*sigmoidf_(x); }
__device__ __forceinline__ float softplusf_(float x){
  return (x > 20.0f) ? x : __logf(1.0f + __expf(x));
}

// ------------------------------------------------------------------
// TDM: issue a 2D tile DMA (global -> LDS).  Tensor is declared as exactly
// the tile (always fully in-bounds here) with the true row stride, so no
// OOB path is exercised.  w0_flags carries data_size=4B + LDS row padding.
// D# packing per CDNA5 ISA 8.3/8.4 (group0: count|lds_addr|global_addr|type,
// group1: flags|dims|tile dims|dim0 stride).
// ------------------------------------------------------------------
__device__ __forceinline__ void tdm_load_2d(unsigned int lds_byte_addr,
                                            const float* gptr,
                                            unsigned int tile_w_elems,
                                            unsigned int tile_h_rows,
                                            unsigned int row_stride_elems,
                                            unsigned int w0_flags)
{
  unsigned long long ga = (unsigned long long)(uintptr_t)gptr;
  u32x4 g0;
  g0.x = 1u;                                                   // count=1, user mode
  g0.y = lds_byte_addr;                                        // [63:32] lds_addr
  g0.z = (unsigned int)ga;                                     // global_addr[31:0]
  g0.w = (unsigned int)((ga >> 32) & 0x01FFFFFFu) | (2u<<30);  // addr[56:32] | type=2
  i32x8 g1;
  g1[0] = (int)w0_flags;                                       // mask=0|dsize|pad cfg
  g1[1] = (int)((tile_w_elems & 0xFFFFu) << 16);               // tensor_dim0 lo
  g1[2] = (int)((tile_h_rows & 0xFFFFu) << 16);                // tdim0 hi | tensor_dim1 lo
  g1[3] = (int)((tile_w_elems & 0xFFFFu) << 16);               // tdim1 hi | tile_dim0
  g1[4] = (int)(tile_h_rows & 0xFFFFu);                        // tile_dim1 | tile_dim2=0
  g1[5] = (int)row_stride_elems;                               // tensor_dim0_stride lo
  g1[6] = 0;
  g1[7] = 0;
  i32x4 z4 = {0,0,0,0};
#if defined(__clang_major__) && (__clang_major__ >= 23)
  i32x8 z8 = {0,0,0,0,0,0,0,0};
  __builtin_amdgcn_tensor_load_to_lds(g0, g1, z4, z4, z8, 0);
#else
  __builtin_amdgcn_tensor_load_to_lds(g0, g1, z4, z4, 0);
#endif
}

// data_size=4B (code 2), pad_enable, pad_interval: 32 dwords (code 4) or
// 64 dwords (code 5), pad_amount: 4 dwords (code 3) -> LDS row strides 36/68.
#define TDM_FLAGS_A ((2u<<16) | (1u<<20) | (4u<<22) | (3u<<25))
#define TDM_FLAGS_B ((2u<<16) | (1u<<20) | (5u<<22) | (3u<<25))

// ------------------------------------------------------------------
// t[b,l,d] = sum_c x[b,c,l] * W_inp[c,d] + b_inp[d]
// ------------------------------------------------------------------
__global__ void k_input_proj(const float* __restrict__ x, const float* __restrict__ Wi,
                             const float* __restrict__ bi, float* __restrict__ t){
  int tid = blockIdx.x*blockDim.x + threadIdx.x;
  if (tid >= BB*LL*DM) return;
  int d  = tid % DM;
  int bl = tid / DM;
  int l  = bl % LL;
  int b  = bl / LL;
  const float* xb = x + (size_t)b*3*LL;
  float acc = bi[d];
  acc += xb[0*LL + l]*Wi[0*DM+d];
  acc += xb[1*LL + l]*Wi[1*DM+d];
  acc += xb[2*LL + l]*Wi[2*DM+d];
  t[tid] = acc;
}

// ------------------------------------------------------------------
// LayerNorm over rows of DM=512; one wave32 per row, 16 elems/lane
// ------------------------------------------------------------------
__global__ void __launch_bounds__(256) k_layernorm(
    const float* __restrict__ x, const float* __restrict__ w,
    const float* __restrict__ b, float* __restrict__ y, int nrows){
  int wave = threadIdx.x >> 5;
  int lane = threadIdx.x & 31;
  int row  = blockIdx.x*(blockDim.x>>5) + wave;
  if (row >= nrows) return;
  const float* xr = x + (size_t)row*DM;
  float v[16];
  float s = 0.f;
  #pragma unroll
  for (int j=0;j<16;j++){ v[j] = xr[lane + j*32]; s += v[j]; }
  #pragma unroll
  for (int m=16;m>=1;m>>=1) s += __shfl_xor(s, m, 32);
  float mu = s * (1.0f/DM);
  float q = 0.f;
  #pragma unroll
  for (int j=0;j<16;j++){ float d = v[j]-mu; q += d*d; }
  #pragma unroll
  for (int m=16;m>=1;m>>=1) q += __shfl_xor(q, m, 32);
  float rs = rsqrtf(q*(1.0f/DM) + LNEPS);
  float* yr = y + (size_t)row*DM;
  #pragma unroll
  for (int j=0;j<16;j++){
    int d = lane + j*32;
    yr[d] = (v[j]-mu)*rs*w[d] + b[d];
  }
}

// ------------------------------------------------------------------
// fp32 WMMA GEMM, TDM-staged + double-buffered LDS.
//   C = act(A[M,K](lda) @ W[K,N](ldw) + bias) (+C if ACCUM)
// 256 thr = 8 waves as 4(M) x 2(N); each wave computes a 32x32 region as
// 2x2 WMMA 16x16 tiles (4 back-to-back v_wmma per k-step, A/B frags reused).
// Block tile 128(M) x 64(N); K chunks of 32 DMA'd into LDS by wave0 (A tile)
// and wave1 (B tile), next chunk issued before computing the current one.
// LDS row strides 36/68 floats produced by the TDM pad feature.
// ------------------------------------------------------------------
#define BMT 128
#define BNT 64
#define KC  32
#define ASTR 36
#define BSTR 68
#define ASZ (BMT*ASTR)     // 4608 floats
#define BSZ (KC*BSTR)      // 2176 floats

template<int ACT, bool ACCUM>
__global__ void __launch_bounds__(256) k_gemm_wmma(
    const float* __restrict__ A, int lda,
    const float* __restrict__ W, int ldw,
    const float* __restrict__ bias,
    float* __restrict__ C, int ldc,
    int M, int N, int K)
{
  __shared__ float As[2*ASZ];
  __shared__ float Bs[2*BSZ];
  const int tid   = threadIdx.x;
  const int waveId= tid >> 5;
  const int lane  = tid & 31;
  const int half  = lane >> 4;
  const int lr    = lane & 15;
  const int waveM = waveId >> 1;   // 0..3 -> 32-row strips
  const int waveN = waveId & 1;    // 0..1 -> 32-col strips
  const int m0 = blockIdx.y * BMT;
  const int n0 = blockIdx.x * BNT;

  const float* Abase = A + (size_t)m0*lda;
  const float* Bbase = W + n0;
  const unsigned int ldsA = (unsigned int)(uintptr_t)&As[0];
  const unsigned int ldsB = (unsigned int)(uintptr_t)&Bs[0];
  const bool issuerA = (waveId == 0);
  const bool issuerB = (waveId == 1);
  const int ns = K / KC;

  v8f acc00 = {0.f,0.f,0.f,0.f,0.f,0.f,0.f,0.f};
  v8f acc01 = acc00, acc10 = acc00, acc11 = acc00;

  // prologue: DMA stage 0 into buffer 0
  if (issuerA) tdm_load_2d(ldsA, Abase, KC, BMT, (unsigned)lda, TDM_FLAGS_A);
  if (issuerB) tdm_load_2d(ldsB, Bbase, BNT, KC, (unsigned)ldw, TDM_FLAGS_B);

  for (int s = 0; s < ns; ++s) {
    const bool more = (s+1 < ns);
    if (more) {
      const int nb = (s+1) & 1;
      if (issuerA) tdm_load_2d(ldsA + nb*ASZ*4, Abase + (size_t)(s+1)*KC,
                               KC, BMT, (unsigned)lda, TDM_FLAGS_A);
      if (issuerB) tdm_load_2d(ldsB + nb*BSZ*4, Bbase + (size_t)(s+1)*KC*ldw,
                               BNT, KC, (unsigned)ldw, TDM_FLAGS_B);
    }
    if (issuerA || issuerB) {
      if (more) __builtin_amdgcn_s_wait_tensorcnt(1);
      else      __builtin_amdgcn_s_wait_tensorcnt(0);
    }
    __syncthreads();

    const float* Ac = &As[(s & 1)*ASZ];
    const float* Bc = &Bs[(s & 1)*BSZ];
    #pragma unroll
    for (int k0=0;k0<KC;k0+=4){
      v2f a0, a1, b0, b1;
      int ai = (waveM*32 + lr)*ASTR + k0 + 2*half;
      a0.x = Ac[ai];            a0.y = Ac[ai+1];
      a1.x = Ac[ai + 16*ASTR];  a1.y = Ac[ai + 16*ASTR + 1];
      int bi = (k0 + 2*half)*BSTR + waveN*32 + lr;
      b0.x = Bc[bi];        b0.y = Bc[bi + BSTR];
      b1.x = Bc[bi + 16];   b1.y = Bc[bi + 16 + BSTR];
      acc00 = __builtin_amdgcn_wmma_f32_16x16x4_f32(false,a0,false,b0,(short)0,acc00,false,false);
      acc01 = __builtin_amdgcn_wmma_f32_16x16x4_f32(false,a0,false,b1,(short)0,acc01,false,false);
      acc10 = __builtin_amdgcn_wmma_f32_16x16x4_f32(false,a1,false,b0,(short)0,acc10,false,false);
      acc11 = __builtin_amdgcn_wmma_f32_16x16x4_f32(false,a1,false,b1,(short)0,acc11,false,false);
    }
    __syncthreads();
  }

  // epilogue: bias + activation + optional residual accumulate
  #pragma unroll
  for (int i=0;i<2;i++){
    #pragma unroll
    for (int j=0;j<2;j++){
      const v8f* accp = (i==0) ? ((j==0)?&acc00:&acc01) : ((j==0)?&acc10:&acc11);
      int col = n0 + waveN*32 + j*16 + lr;
      float bv = bias ? bias[col] : 0.0f;
      #pragma unroll
      for (int r=0;r<8;r++){
        int row = m0 + waveM*32 + i*16 + r + half*8;
        float v = (*accp)[r] + bv;
        if (ACT==1) v = softplusf_(v);
        size_t ci = (size_t)row*ldc + col;
        if (ACCUM) v += C[ci];
        C[ci] = v;
      }
    }
  }
}

// ------------------------------------------------------------------
// Depthwise causal conv (D_CONV=4) + SiLU.  u = xz[..., 0:DI] (stride 2*DI)
// ------------------------------------------------------------------
__global__ void k_conv_silu(const float* __restrict__ xz, const float* __restrict__ cw,
                            const float* __restrict__ cb, float* __restrict__ uc){
  int tid = blockIdx.x*blockDim.x + threadIdx.x;
  if (tid >= BB*LL*DI) return;
  int e  = tid % DI;
  int bl = tid / DI;
  int l  = bl % LL;
  int b  = bl / LL;
  const float* u = xz + (size_t)b*LL*2*DI;
  float acc = cb[e];
  #pragma unroll
  for (int k=0;k<4;k++){
    int ls = l + k - 3;
    float uv = (ls >= 0) ? u[(size_t)ls*2*DI + e] : 0.f;
    acc += uv * cw[e*4 + k];
  }
  uc[tid] = siluf_(acc);
}

// ------------------------------------------------------------------
// Selective scan with Luenberger correction. Thread = (b,e), 16 states in
// registers; B_t/C_t broadcast via LDS. Fuses D-skip + SiLU(z) gating and
// writes the gated y IN PLACE over uc (read-before-write per element).
// ------------------------------------------------------------------
__global__ void __launch_bounds__(256) k_scan(
    const float* __restrict__ ucy,
    const float* __restrict__ dtb,
    const float* __restrict__ dbc,
    const float* __restrict__ xz,     // z = xz[..., DI:2*DI]
    const float* __restrict__ Alog,
    const float* __restrict__ Dskip,
    float* __restrict__ yout)         // == ucy
{
  int b = blockIdx.y;
  int e = blockIdx.x*blockDim.x + threadIdx.x;
  float A[DS], h[DS];
  #pragma unroll
  for (int n=0;n<DS;n++){ A[n] = -__expf(Alog[e*DS+n]); h[n]=0.f; }
  float Dv = Dskip[e];
  __shared__ float sB[DS], sC[DS];
  for (int t=0;t<LL;t++){
    size_t base = (size_t)(b*LL + t);
    if (threadIdx.x < 2*DS) {
      float v = dbc[base*64 + DTR + threadIdx.x];
      if (threadIdx.x < DS) sB[threadIdx.x] = v; else sC[threadIdx.x-DS] = v;
    }
    __syncthreads();
    float dtv = dtb[base*DI + e];
    float uv  = ucy[base*DI + e];
    float zv  = xz[base*2*DI + DI + e];
    float du  = dtv*uv;
    float err = uv;
    #pragma unroll
    for (int n=0;n<DS;n++){
      h[n] = __expf(dtv*A[n])*h[n] + du*sB[n];
      err -= h[n]*sC[n];
    }
    float ea = ALPHA_C*err;
    float yv = 0.f;
    #pragma unroll
    for (int n=0;n<DS;n++){
      h[n] += ea*sC[n];
      yv   += h[n]*sC[n];
    }
    float yo = (yv + Dv*uv) * siluf_(zv);
    yout[base*DI + e] = yo;
    __syncthreads();
  }
}

// ------------------------------------------------------------------
// mean over L -> final LN -> classifier. One block per batch element.
// ------------------------------------------------------------------
__global__ void __launch_bounds__(512) k_head(
    const float* __restrict__ t, const float* __restrict__ lnw, const float* __restrict__ lnb,
    const float* __restrict__ Wc, const float* __restrict__ bc, float* __restrict__ out)
{
  int b = blockIdx.x;
  int d = threadIdx.x;            // 0..511
  __shared__ float sv[DM];
  __shared__ float red[DM];
  float s = 0.f;
  for (int l=0;l<LL;l++) s += t[((size_t)(b*LL+l))*DM + d];
  float p = s * (1.0f/LL);
  red[d] = p; __syncthreads();
  for (int off=256; off>=1; off>>=1){ if (d < off) red[d] += red[d+off]; __syncthreads(); }
  float mu = red[0] * (1.0f/DM);
  __syncthreads();
  float dd = p - mu;
  red[d] = dd*dd; __syncthreads();
  for (int off=256; off>=1; off>>=1){ if (d < off) red[d] += red[d+off]; __syncthreads(); }
  float rs = rsqrtf(red[0]*(1.0f/DM) + LNEPS);
  sv[d] = dd*rs*lnw[d] + lnb[d];
  __syncthreads();
  if (d < NC){
    float acc = bc[d];
    for (int k=0;k<DM;k++) acc += sv[k]*Wc[k*NC + d];
    out[b*NC + d] = acc;
  }
}

// ------------------------------------------------------------------
extern "C" void kernel_launch(void* const* d_in, const int* in_sizes, int n_in,
                              void* d_out, int out_size, void* d_ws, size_t ws_size,
                              hipStream_t stream)
{
  (void)in_sizes; (void)n_in; (void)out_size; (void)ws_size;
  const float* x     = (const float*)d_in[0];
  const float* W_inp = (const float*)d_in[1];
  const float* b_inp = (const float*)d_in[2];
  const float* W_in  = (const float*)d_in[3];
  const float* b_in  = (const float*)d_in[4];
  const float* conv_w= (const float*)d_in[5];
  const float* conv_b= (const float*)d_in[6];
  const float* W_x   = (const float*)d_in[7];
  const float* W_dt  = (const float*)d_in[8];
  const float* b_dt  = (const float*)d_in[9];
  const float* A_log = (const float*)d_in[10];
  const float* D_skip= (const float*)d_in[11];
  const float* W_out = (const float*)d_in[12];
  const float* b_out = (const float*)d_in[13];
  const float* ln_w  = (const float*)d_in[14];
  const float* ln_b  = (const float*)d_in[15];
  const float* lnf_w = (const float*)d_in[16];
  const float* lnf_b = (const float*)d_in[17];
  const float* W_cls = (const float*)d_in[18];
  const float* b_cls = (const float*)d_in[19];
  float* out = (float*)d_out;

  float* ws = (float*)d_ws;
  size_t o = 0;
  float* t   = ws + o;  o += (size_t)BB*LL*DM;     // residual stream  16 MB
  float* ln  = ws + o;  o += (size_t)BB*LL*DM;     // LN output        16 MB
  float* xz  = ws + o;  o += (size_t)BB*LL*2*DI;   // u | z            64 MB
  float* uc  = ws + o;  o += (size_t)BB*LL*DI;     // conv out / y     32 MB
  float* dbc = ws + o;  o += (size_t)BB*LL*64;     // dt_r|B|C          2 MB
  float* dtb = ws + o;  o += (size_t)BB*LL*DI;     // dt               32 MB

  const int M = BB*LL;   // 8192

  k_input_proj<<<(BB*LL*DM+255)/256, 256, 0, stream>>>(x, W_inp, b_inp, t);

  for (int i=0;i<NL;i++){
    k_layernorm<<<M/8, 256, 0, stream>>>(t, ln_w + i*DM, ln_b + i*DM, ln, M);

    // xz = ln @ W_in[i] + b_in[i]        (8192 x 512 x 2048)
    k_gemm_wmma<0,false><<<dim3(2*DI/BNT, M/BMT), 256, 0, stream>>>(
        ln, DM, W_in + (size_t)i*DM*2*DI, 2*DI, b_in + (size_t)i*2*DI, xz, 2*DI, M, 2*DI, DM);

    k_conv_silu<<<(M*DI+255)/256, 256, 0, stream>>>(
        xz, conv_w + (size_t)i*DI*4, conv_b + (size_t)i*DI, uc);

    // dbc = uc @ W_x[i]                  (8192 x 1024 x 64, no bias)
    k_gemm_wmma<0,false><<<dim3(64/BNT, M/BMT), 256, 0, stream>>>(
        uc, DI, W_x + (size_t)i*DI*64, 64, nullptr, dbc, 64, M, 64, DI);

    // dt = softplus(dbc[:, :32] @ W_dt[i] + b_dt[i])   (8192 x 32 x 1024)
    k_gemm_wmma<1,false><<<dim3(DI/BNT, M/BMT), 256, 0, stream>>>(
        dbc, 64, W_dt + (size_t)i*DTR*DI, DI, b_dt + (size_t)i*DI, dtb, DI, M, DI, DTR);

    // sequential scan; writes gated y in place over uc
    k_scan<<<dim3(DI/256, BB), 256, 0, stream>>>(
        uc, dtb, dbc, xz, A_log + (size_t)i*DI*DS, D_skip + (size_t)i*DI, uc);

    // t += y @ W_out[i] + b_out[i]       (8192 x 1024 x 512, accumulate)
    k_gemm_wmma<0,true><<<dim3(DM/BNT, M/BMT), 256, 0, stream>>>(
        uc, DI, W_out + (size_t)i*DI*DM, DM, b_out + (size_t)i*DM, t, DM, M, DM, DI);
  }

  k_head<<<BB, 512, 0, stream>>>(t, lnf_w, lnf_b, W_cls, b_cls, out);
}